// DTI_PU_loss_58162447123174
// MI455X (gfx1250) — compile-verified
//
#include <hip/hip_runtime.h>

// Problem constants from the reference: matrices are 8192 x 8192 f32.
// y < 8192 and x < 8192  =>  flat offset = (x << 13) + y  < 2^26, and the
// byte offset < 2^28: keep it in 32-bit unsigned so the compiler emits the
// saddr + 32-bit-voffset global-load form (1 VGPR per gather address).
#define MSHIFT 13

typedef int vint4 __attribute__((ext_vector_type(4)));

// Fused pass over BOTH index-pair streams (pos handled by blocks [0, split),
// neg by blocks [split, gridDim.x)): acc += (recon[off] - dp[off])^2.
//  - Index streams are read exactly once -> non-temporal b128 loads (TH=NT)
//    so 80 MB of indices don't evict matrix lines from the 192 MB L2, which
//    is capturing the ~5x cacheline reuse of the gathered matrices (RT hint).
//  - 8 items per thread per grid-stride step: 4x b128 NT index loads, then
//    16 independent b32 gathers in flight per lane -> MLP hides random HBM
//    latency (the binding constraint at 23.3 TB/s).
//  - wave32 shuffle tree + LDS block reduction; one deterministic f32
//    partial per block (no float atomics -> bit-stable across graph replays).
__global__ __launch_bounds__(256) void gather_sq_partial(
    const float* __restrict__ recon, const float* __restrict__ dp,
    const int* __restrict__ x0, const int* __restrict__ y0, int n0,
    const int* __restrict__ x1, const int* __restrict__ y1, int n1,
    int split, float* __restrict__ partial)
{
    const int* xi; const int* yi; int n, lblk, lgrid;
    if ((int)blockIdx.x < split) {
        xi = x0; yi = y0; n = n0; lblk = blockIdx.x;        lgrid = split;
    } else {
        xi = x1; yi = y1; n = n1; lblk = blockIdx.x - split; lgrid = gridDim.x - split;
    }

    float acc = 0.0f;
    int tid    = lblk * blockDim.x + threadIdx.x;
    int stride = lgrid * blockDim.x;

    int nocts = n >> 3;  // groups of 8
    for (int q = tid; q < nocts; q += stride) {
        vint4 xa = __builtin_nontemporal_load((const vint4*)xi + 2 * q);
        vint4 xb = __builtin_nontemporal_load((const vint4*)xi + 2 * q + 1);
        vint4 ya = __builtin_nontemporal_load((const vint4*)yi + 2 * q);
        vint4 yb = __builtin_nontemporal_load((const vint4*)yi + 2 * q + 1);
        float d0[8], d1[8];
#pragma unroll
        for (int j = 0; j < 4; ++j) {
            unsigned offa = ((unsigned)xa[j] << MSHIFT) + (unsigned)ya[j];
            unsigned offb = ((unsigned)xb[j] << MSHIFT) + (unsigned)yb[j];
            d0[j]     = recon[offa]; d1[j]     = dp[offa];
            d0[j + 4] = recon[offb]; d1[j + 4] = dp[offb];
        }
#pragma unroll
        for (int j = 0; j < 8; ++j) {
            float d = d0[j] - d1[j];
            acc = fmaf(d, d, acc);
        }
    }
    // Tail (n % 8 leftovers).
    for (int i = (nocts << 3) + tid; i < n; i += stride) {
        unsigned off = ((unsigned)xi[i] << MSHIFT) + (unsigned)yi[i];
        float d = recon[off] - dp[off];
        acc = fmaf(d, d, acc);
    }

    // Wave32 reduction (CDNA5 is wave32-only; never assume 64 lanes).
#pragma unroll
    for (int o = 16; o > 0; o >>= 1) acc += __shfl_down(acc, o, 32);

    __shared__ float wsum[8];  // 256 threads = 8 wave32 waves
    int lane = threadIdx.x & 31;
    int wave = threadIdx.x >> 5;
    if (lane == 0) wsum[wave] = acc;
    __syncthreads();
    if (wave == 0) {
        float v = (lane < 8) ? wsum[lane] : 0.0f;
#pragma unroll
        for (int o = 4; o > 0; o >>= 1) v += __shfl_down(v, o, 32);
        if (lane == 0) partial[blockIdx.x] = v;
    }
}

// Sum partial[0..split) -> pos, partial[split..total) -> neg, apply alpha.
__global__ __launch_bounds__(256) void combine_partials(
    const float* __restrict__ partial, int split, int total,
    const float* __restrict__ alpha, float* __restrict__ out)
{
    float ps = 0.0f, ns = 0.0f;
    for (int i = threadIdx.x; i < total; i += blockDim.x) {
        float v = partial[i];
        if (i < split) ps += v; else ns += v;
    }
#pragma unroll
    for (int o = 16; o > 0; o >>= 1) {
        ps += __shfl_down(ps, o, 32);
        ns += __shfl_down(ns, o, 32);
    }
    __shared__ float sp[8], sn[8];
    int lane = threadIdx.x & 31;
    int wave = threadIdx.x >> 5;
    if (lane == 0) { sp[wave] = ps; sn[wave] = ns; }
    __syncthreads();
    if (threadIdx.x == 0) {
        float tps = 0.0f, tns = 0.0f;
#pragma unroll
        for (int w = 0; w < 8; ++w) { tps += sp[w]; tns += sn[w]; }
        float a = alpha[0];
        out[0] = tps * ((1.0f - a) * 0.5f) + tns * (a * 0.5f);
    }
}

extern "C" void kernel_launch(void* const* d_in, const int* in_sizes, int n_in,
                              void* d_out, int out_size, void* d_ws, size_t ws_size,
                              hipStream_t stream) {
    // setup_inputs() order:
    // 0: drug_protein_reconstruct (f32, 8192*8192)
    // 1: drug_protein             (f32, 8192*8192)
    // 2: alpha                    (f32, 1)
    // 3: pos_x_index (i32)  4: pos_y_index (i32)
    // 5: neg_x_index (i32)  6: neg_y_index (i32)
    const float* recon = (const float*)d_in[0];
    const float* dp    = (const float*)d_in[1];
    const float* alpha = (const float*)d_in[2];
    const int* pos_x   = (const int*)d_in[3];
    const int* pos_y   = (const int*)d_in[4];
    const int* neg_x   = (const int*)d_in[5];
    const int* neg_y   = (const int*)d_in[6];
    const int n_pos = in_sizes[3];
    const int n_neg = in_sizes[5];

    // One fused launch: 2560 blocks x 8 wave32 waves = 20K waves of random
    // gathers in flight; blocks apportioned by stream length so pos and neg
    // finish together. Partials need only total_blocks * 4 B of workspace.
    int total = 2560;
    if (ws_size < (size_t)total * sizeof(float)) {
        total = (int)(ws_size / sizeof(float));
        if (total < 2) total = 2;
    }
    long long np = n_pos, nn = n_neg;
    int split = (int)((long long)total * np / (np + nn > 0 ? np + nn : 1));
    if (split < 1) split = 1;
    if (split > total - 1) split = total - 1;

    float* partial = (float*)d_ws;

    gather_sq_partial<<<total, 256, 0, stream>>>(
        recon, dp, pos_x, pos_y, n_pos, neg_x, neg_y, n_neg, split, partial);
    combine_partials<<<1, 256, 0, stream>>>(partial, split, total, alpha, (float*)d_out);
}